// GAT_15925738733669
// MI455X (gfx1250) — compile-verified
//
#include <hip/hip_runtime.h>
#include <hip/hip_bf16.h>
#include <math.h>
#include <stdint.h>

// ---------------------------------------------------------------------------
// GAT 2-layer forward for MI455X (gfx1250, wave32).
// GEMMs use native fp32 WMMA (V_WMMA_F32_16X16X4_F32), LDS-tiled with
// compile-time strides so the inner loop is pure ds_load_b64 + v_wmma.
// Edge softmax/aggregation uses f32 global atomics (L2-resident working set).
// ---------------------------------------------------------------------------

typedef float v2f __attribute__((ext_vector_type(2)));
typedef float v8f __attribute__((ext_vector_type(8)));

#define NEG_SLOPE 0.2f

__device__ __forceinline__ void atomicMaxF32(float* addr, float val) {
  // Sign-split trick: monotone int max for >=0, monotone unsigned min for <0.
  if (val >= 0.0f) {
    atomicMax((int*)addr, __float_as_int(val));
  } else {
    atomicMin((unsigned int*)addr, __float_as_uint(val));
  }
}

// ---------------------------------------------------------------------------
// C[M x NACT] = A[M x 128] * B[128 x NACT]
// grid = (M/16, ceil(NACT/(16*NTILES))), blockDim = 32*NTILES.
// Block stages a 16x128 A-strip and a transposed, zero-padded B panel in LDS.
// Wave w computes the 16x16 tile at local columns [16w, 16w+16).
// Fragment layouts per CDNA5 ISA 7.12.2:
//   A (16x4, MxK): lanes 0-15 -> M=lane, VGPR0/1 = K0/K1; lanes 16-31 -> K2/K3
//   B (4x16, KxN): lanes 0-15 -> N=lane, VGPR0/1 = K0/K1; lanes 16-31 -> K2/K3
//   C (16x16):     VGPR r -> M = r + (lane>=16 ? 8 : 0); N = lane & 15
// ---------------------------------------------------------------------------
template <int NTILES, int NACT>
__global__ void __launch_bounds__(32 * NTILES)
gat_wmma_gemm_k128(const float* __restrict__ A, const float* __restrict__ B,
                   float* __restrict__ C) {
  constexpr int NPAD = NTILES * 16;   // columns staged per block
  constexpr int SA = 132;             // LDS row stride (132 % 64 == 4 banks)
  constexpr int SB = 132;
  __shared__ float ldsA[16 * SA];
  __shared__ float ldsB[NPAD * SB];

  const int tid  = threadIdx.x;
  const int lane = tid & 31;
  const int wave = tid >> 5;
  const size_t m0 = (size_t)blockIdx.x * 16;
  const int c0 = blockIdx.y * NPAD;   // global column offset of this panel

  // Stage A strip (16 x 128), coalesced.
  for (int i = tid; i < 16 * 128; i += 32 * NTILES) {
    int r = i >> 7, c = i & 127;
    ldsA[r * SA + c] = A[(m0 + r) * 128 + c];
  }
  // Stage B panel transposed + zero-padded: ldsB[n][k] = B[k][c0+n].
  for (int i = tid; i < NPAD * 128; i += 32 * NTILES) {
    int k = i / NPAD, n = i - k * NPAD;
    int gn = c0 + n;
    ldsB[n * SB + k] = (gn < NACT) ? B[(size_t)k * NACT + gn] : 0.0f;
  }
  __syncthreads();

  const int l15  = lane & 15;
  const int koff = (lane >> 4) << 1;          // 0 | 2
  const int ncol = wave * 16 + l15;           // local column
  const float* ap = &ldsA[l15 * SA + koff];
  const float* bp = &ldsB[ncol * SB + koff];

  v8f acc = {0.f, 0.f, 0.f, 0.f, 0.f, 0.f, 0.f, 0.f};
#pragma unroll
  for (int kb = 0; kb < 128; kb += 4) {
    v2f a = *(const v2f*)(ap + kb);           // ds_load_b64, immediate offset
    v2f b = *(const v2f*)(bp + kb);           // ds_load_b64, immediate offset
    acc = __builtin_amdgcn_wmma_f32_16x16x4_f32(
        /*neg_a=*/false, a, /*neg_b=*/false, b,
        /*c_mod=*/(short)0, acc, /*reuse_a=*/false, /*reuse_b=*/false);
  }

  const int gcol = c0 + ncol;
  if (gcol < NACT) {
    float* cp = C + (m0 + (size_t)((lane >> 4) << 3)) * NACT + gcol;
#pragma unroll
    for (int r = 0; r < 8; ++r) cp[(size_t)r * NACT] = acc[r];
  }
}

// ---------------------------------------------------------------------------
__global__ void fill_kernel(float* __restrict__ p, float v, long long n) {
  long long i = (long long)blockIdx.x * blockDim.x + threadIdx.x;
  long long stride = (long long)gridDim.x * blockDim.x;
  for (; i < n; i += stride) p[i] = v;
}

__global__ void init_out_kernel(float* __restrict__ out, const float* __restrict__ b2,
                                int Nn) {
  int idx = blockIdx.x * blockDim.x + threadIdx.x;
  if (idx >= Nn * 40) return;
  out[idx] = b2[idx % 40];
}

// a_src/a_dst for layer 1: h is [N, 8, 16], att is [8,16]
__global__ void scores1_kernel(const float* __restrict__ h,
                               const float* __restrict__ att_src,
                               const float* __restrict__ att_dst,
                               float* __restrict__ a_src, float* __restrict__ a_dst,
                               int Nn) {
  int idx = blockIdx.x * blockDim.x + threadIdx.x;
  if (idx >= Nn * 8) return;
  int n = idx >> 3, hd = idx & 7;
  const float* hp = h + (size_t)n * 128 + hd * 16;
  float s = 0.f, d = 0.f;
#pragma unroll
  for (int c = 0; c < 16; ++c) {
    float v = hp[c];
    s = fmaf(v, att_src[hd * 16 + c], s);
    d = fmaf(v, att_dst[hd * 16 + c], d);
  }
  a_src[idx] = s;
  a_dst[idx] = d;
}

// layer 2: h2 is [N, 40], att is [1,40]
__global__ void scores2_kernel(const float* __restrict__ h2,
                               const float* __restrict__ att_src,
                               const float* __restrict__ att_dst,
                               float* __restrict__ a_src, float* __restrict__ a_dst,
                               int Nn) {
  int n = blockIdx.x * blockDim.x + threadIdx.x;
  if (n >= Nn) return;
  const float* hp = h2 + (size_t)n * 40;
  float s = 0.f, d = 0.f;
#pragma unroll
  for (int c = 0; c < 40; ++c) {
    float v = hp[c];
    s = fmaf(v, att_src[c], s);
    d = fmaf(v, att_dst[c], d);
  }
  a_src[n] = s;
  a_dst[n] = d;
}

// ------------------------- layer-1 edge passes (H = 8) ---------------------
__global__ void edge_max1_kernel(const int* __restrict__ src, const int* __restrict__ dst,
                                 const float* __restrict__ a_src, const float* __restrict__ a_dst,
                                 float* __restrict__ m, int E, int Etot) {
  int idx = blockIdx.x * blockDim.x + threadIdx.x;
  if (idx >= Etot * 8) return;
  int e = idx >> 3, hd = idx & 7;
  int s = (e < E) ? src[e] : (e - E);
  int d = (e < E) ? dst[e] : (e - E);
  float v = a_src[s * 8 + hd] + a_dst[d * 8 + hd];
  v = (v >= 0.f) ? v : NEG_SLOPE * v;
  atomicMaxF32(&m[d * 8 + hd], v);
}

__global__ void edge_sum1_kernel(const int* __restrict__ src, const int* __restrict__ dst,
                                 const float* __restrict__ a_src, const float* __restrict__ a_dst,
                                 const float* __restrict__ m, float* __restrict__ ssum,
                                 int E, int Etot) {
  int idx = blockIdx.x * blockDim.x + threadIdx.x;
  if (idx >= Etot * 8) return;
  int e = idx >> 3, hd = idx & 7;
  int s = (e < E) ? src[e] : (e - E);
  int d = (e < E) ? dst[e] : (e - E);
  float v = a_src[s * 8 + hd] + a_dst[d * 8 + hd];
  v = (v >= 0.f) ? v : NEG_SLOPE * v;
  atomicAdd(&ssum[d * 8 + hd], __expf(v - m[d * 8 + hd]));
}

// one wave32 per edge; lane owns 4 consecutive features (head = lane>>2)
__global__ void edge_agg1_kernel(const int* __restrict__ src, const int* __restrict__ dst,
                                 const float* __restrict__ a_src, const float* __restrict__ a_dst,
                                 const float* __restrict__ m, const float* __restrict__ ssum,
                                 const float* __restrict__ h, float* __restrict__ agg,
                                 int E, int Etot) {
  int wid  = (int)(((long long)blockIdx.x * blockDim.x + threadIdx.x) >> 5);
  int lane = threadIdx.x & 31;
  if (wid >= Etot) return;
  int s = (wid < E) ? src[wid] : (wid - E);
  int d = (wid < E) ? dst[wid] : (wid - E);
  int hd = lane >> 2;
  float e = a_src[s * 8 + hd] + a_dst[d * 8 + hd];
  e = (e >= 0.f) ? e : NEG_SLOPE * e;
  float alpha = __expf(e - m[d * 8 + hd]) / ssum[d * 8 + hd];
  const float4 hv = *(const float4*)(h + (size_t)s * 128 + lane * 4);
  float* op = agg + (size_t)d * 128 + lane * 4;
  atomicAdd(op + 0, hv.x * alpha);
  atomicAdd(op + 1, hv.y * alpha);
  atomicAdd(op + 2, hv.z * alpha);
  atomicAdd(op + 3, hv.w * alpha);
}

__global__ void elu_bias_kernel(float* __restrict__ a, const float* __restrict__ b,
                                long long n) {
  long long idx = (long long)blockIdx.x * blockDim.x + threadIdx.x;
  if (idx >= n) return;
  float v = a[idx] + b[idx & 127];
  a[idx] = (v > 0.f) ? v : (__expf(v) - 1.0f);
}

// ------------------------- layer-2 edge passes (H = 1) ---------------------
__global__ void edge_max2_kernel(const int* __restrict__ src, const int* __restrict__ dst,
                                 const float* __restrict__ a_src, const float* __restrict__ a_dst,
                                 float* __restrict__ m, int E, int Etot) {
  int e = blockIdx.x * blockDim.x + threadIdx.x;
  if (e >= Etot) return;
  int s = (e < E) ? src[e] : (e - E);
  int d = (e < E) ? dst[e] : (e - E);
  float v = a_src[s] + a_dst[d];
  v = (v >= 0.f) ? v : NEG_SLOPE * v;
  atomicMaxF32(&m[d], v);
}

__global__ void edge_sum2_kernel(const int* __restrict__ src, const int* __restrict__ dst,
                                 const float* __restrict__ a_src, const float* __restrict__ a_dst,
                                 const float* __restrict__ m, float* __restrict__ ssum,
                                 int E, int Etot) {
  int e = blockIdx.x * blockDim.x + threadIdx.x;
  if (e >= Etot) return;
  int s = (e < E) ? src[e] : (e - E);
  int d = (e < E) ? dst[e] : (e - E);
  float v = a_src[s] + a_dst[d];
  v = (v >= 0.f) ? v : NEG_SLOPE * v;
  atomicAdd(&ssum[d], __expf(v - m[d]));
}

// one wave32 per edge, 40 features
__global__ void edge_agg2_kernel(const int* __restrict__ src, const int* __restrict__ dst,
                                 const float* __restrict__ a_src, const float* __restrict__ a_dst,
                                 const float* __restrict__ m, const float* __restrict__ ssum,
                                 const float* __restrict__ h2, float* __restrict__ out,
                                 int E, int Etot) {
  int wid  = (int)(((long long)blockIdx.x * blockDim.x + threadIdx.x) >> 5);
  int lane = threadIdx.x & 31;
  if (wid >= Etot) return;
  int s = (wid < E) ? src[wid] : (wid - E);
  int d = (wid < E) ? dst[wid] : (wid - E);
  float e = a_src[s] + a_dst[d];
  e = (e >= 0.f) ? e : NEG_SLOPE * e;
  float alpha = __expf(e - m[d]) / ssum[d];
  for (int c = lane; c < 40; c += 32)
    atomicAdd(&out[(size_t)d * 40 + c], h2[(size_t)s * 40 + c] * alpha);
}

// ---------------------------------------------------------------------------
extern "C" void kernel_launch(void* const* d_in, const int* in_sizes, int n_in,
                              void* d_out, int out_size, void* d_ws, size_t ws_size,
                              hipStream_t stream) {
  const float* x   = (const float*)d_in[0];
  const int*   ei  = (const int*)d_in[1];
  const float* W1  = (const float*)d_in[2];
  const float* as1 = (const float*)d_in[3];
  const float* ad1 = (const float*)d_in[4];
  const float* b1  = (const float*)d_in[5];
  const float* W2  = (const float*)d_in[6];
  const float* as2 = (const float*)d_in[7];
  const float* ad2 = (const float*)d_in[8];
  const float* b2  = (const float*)d_in[9];
  float* out = (float*)d_out;

  const int Nn   = in_sizes[0] / 128;  // 100000
  const int E    = in_sizes[1] / 2;    // 1600000
  const int Etot = E + Nn;             // self-loops appended
  const int* srcp = ei;
  const int* dstp = ei + E;

  // workspace layout (floats)
  float* ws   = (float*)d_ws;
  float* h    = ws;  ws += (size_t)Nn * 128;   // layer-1 linear output
  float* agg1 = ws;  ws += (size_t)Nn * 128;   // layer-1 aggregate -> elu in place
  float* asc1 = ws;  ws += (size_t)Nn * 8;
  float* adc1 = ws;  ws += (size_t)Nn * 8;
  float* m1   = ws;  ws += (size_t)Nn * 8;
  float* s1   = ws;  ws += (size_t)Nn * 8;
  float* h2   = ws;  ws += (size_t)Nn * 40;    // layer-2 linear output
  float* asc2 = ws;  ws += Nn;
  float* adc2 = ws;  ws += Nn;
  float* m2   = ws;  ws += Nn;
  float* s2   = ws;  ws += Nn;

  const int T = 256;
  const int fillBlocks = 2048;

  // ---- init accumulators (every call: replay-safe, deterministic work) ----
  fill_kernel<<<fillBlocks, T, 0, stream>>>(m1, -INFINITY, (long long)Nn * 8);
  fill_kernel<<<fillBlocks, T, 0, stream>>>(s1, 0.0f, (long long)Nn * 8);
  fill_kernel<<<fillBlocks, T, 0, stream>>>(agg1, 0.0f, (long long)Nn * 128);
  fill_kernel<<<fillBlocks, T, 0, stream>>>(m2, -INFINITY, (long long)Nn);
  fill_kernel<<<fillBlocks, T, 0, stream>>>(s2, 0.0f, (long long)Nn);
  init_out_kernel<<<(Nn * 40 + T - 1) / T, T, 0, stream>>>(out, b2, Nn);

  // ---- layer 1: h = x @ W1 (N=128 split into two 64-col panels) ----
  gat_wmma_gemm_k128<4, 128><<<dim3(Nn / 16, 2), 4 * 32, 0, stream>>>(x, W1, h);
  scores1_kernel<<<(Nn * 8 + T - 1) / T, T, 0, stream>>>(h, as1, ad1, asc1, adc1, Nn);
  edge_max1_kernel<<<(Etot * 8 + T - 1) / T, T, 0, stream>>>(srcp, dstp, asc1, adc1, m1, E, Etot);
  edge_sum1_kernel<<<(Etot * 8 + T - 1) / T, T, 0, stream>>>(srcp, dstp, asc1, adc1, m1, s1, E, Etot);
  edge_agg1_kernel<<<(Etot + 7) / 8, T, 0, stream>>>(srcp, dstp, asc1, adc1, m1, s1, h, agg1, E, Etot);
  elu_bias_kernel<<<(int)(((long long)Nn * 128 + T - 1) / T), T, 0, stream>>>(agg1, b1, (long long)Nn * 128);

  // ---- layer 2: h2 = elu(h1) @ W2 (N=40, one 48-col zero-padded panel) ----
  gat_wmma_gemm_k128<3, 40><<<dim3(Nn / 16, 1), 3 * 32, 0, stream>>>(agg1, W2, h2);
  scores2_kernel<<<(Nn + T - 1) / T, T, 0, stream>>>(h2, as2, ad2, asc2, adc2, Nn);
  edge_max2_kernel<<<(Etot + T - 1) / T, T, 0, stream>>>(srcp, dstp, asc2, adc2, m2, E, Etot);
  edge_sum2_kernel<<<(Etot + T - 1) / T, T, 0, stream>>>(srcp, dstp, asc2, adc2, m2, s2, E, Etot);
  edge_agg2_kernel<<<(Etot + 7) / 8, T, 0, stream>>>(srcp, dstp, asc2, adc2, m2, s2, h2, out, E, Etot);
}